// OldSineGen_47656957116656
// MI455X (gfx1250) — compile-verified
//
#include <hip/hip_runtime.h>
#include <math.h>

#define SR_F     24000.0f
#define TWO_PI_F 6.28318530717958647692f
#define T_LEN    240000
#define NB       8
#define NH       9
#define CHUNK    2048
#define NCHUNK   118          // ceil(240000 / 2048)
#define THREADS  256
#define PT       8            // elements per thread (T_LEN % PT == 0)

typedef float v2f __attribute__((ext_vector_type(2)));
typedef float v8f __attribute__((ext_vector_type(8)));

// CDNA5 async global->LDS copy (ASYNCcnt-tracked). Inline asm: portable across
// ROCm-7.2 / amdgpu-toolchain builtin arity differences.
__device__ __forceinline__ void async_load_b128(unsigned lds_byte_addr, const float* gsrc) {
  asm volatile("global_load_async_to_lds_b128 %0, %1, off"
               :: "v"(lds_byte_addr), "v"(gsrc)
               : "memory");
}

// ---------------- Kernel 1: per-chunk partial sums of f0 ----------------
__global__ __launch_bounds__(THREADS)
void k_chunk_sums(const float* __restrict__ f0, float* __restrict__ sums) {
  const int c = blockIdx.x, b = blockIdx.y;
  const int t = threadIdx.x;
  const int e = c * CHUNK + t * PT;          // time index within batch
  float s = 0.f;
  if (e < T_LEN) {
    const float4* p = (const float4*)(f0 + (size_t)b * T_LEN + e);
    float4 x = p[0], y = p[1];
    s = ((x.x + x.y) + (x.z + x.w)) + ((y.x + y.y) + (y.z + y.w));
  }
  // wave32 reduction
  #pragma unroll
  for (int off = 16; off > 0; off >>= 1) s += __shfl_down(s, off, 32);
  __shared__ float ws[THREADS / 32];
  if ((t & 31) == 0) ws[t >> 5] = s;
  __syncthreads();
  if (t == 0) {
    float tot = 0.f;
    #pragma unroll
    for (int w = 0; w < THREADS / 32; ++w) tot += ws[w];
    sums[b * NCHUNK + c] = tot;
  }
}

// ---- Kernel 2: exclusive scan of chunk sums (one wave32 per batch) -----
__global__ __launch_bounds__(NB * 32)
void k_scan(float* sums) {
  const int w    = threadIdx.x >> 5;    // batch index
  const int lane = threadIdx.x & 31;
  float carry = 0.f;
  #pragma unroll
  for (int r = 0; r < (NCHUNK + 31) / 32; ++r) {
    const int c = r * 32 + lane;
    const bool in = (c < NCHUNK);
    float x = in ? sums[w * NCHUNK + c] : 0.f;
    float incl = x;
    #pragma unroll
    for (int off = 1; off < 32; off <<= 1) {   // wave32 inclusive shuffle scan
      float y = __shfl_up(incl, off, 32);
      if (lane >= off) incl += y;
    }
    if (in) sums[w * NCHUNK + c] = carry + (incl - x);  // exclusive prefix
    carry += __shfl(incl, 31, 32);
  }
}

// ---------------- Kernel 3: scan + 9 harmonics of sin -------------------
__global__ __launch_bounds__(THREADS)
void k_main(const float* __restrict__ f0, const float* __restrict__ jit,
            const float* __restrict__ coff, float* __restrict__ out) {
  const int c = blockIdx.x, b = blockIdx.y;
  const int t = threadIdx.x;
  const int e = c * CHUNK + t * PT;
  const bool valid = e < T_LEN;              // all-or-nothing per thread (T%PT==0)

  __shared__ float totals[THREADS];
  __shared__ float scanD[THREADS];
  __shared__ float colOff[16];
  __shared__ float jbuf[2][CHUNK];

  // Load this thread's 8 contiguous f0 values; sequential inclusive scan.
  float v[PT];
  if (valid) {
    const float4* p = (const float4*)(f0 + (size_t)b * T_LEN + e);
    float4 x = p[0], y = p[1];
    v[0] = x.x; v[1] = x.y; v[2] = x.z; v[3] = x.w;
    v[4] = y.x; v[5] = y.y; v[6] = y.z; v[7] = y.w;
  } else {
    #pragma unroll
    for (int i = 0; i < PT; ++i) v[i] = 0.f;
  }
  #pragma unroll
  for (int i = 1; i < PT; ++i) v[i] += v[i - 1];
  totals[t] = v[PT - 1];

  // Two-deep async prefetch of jitter (h=0 -> buf0, h=1 -> buf1). The ~2KB/wave
  // in flight overlaps the whole scan section below. Each lane later consumes
  // exactly the LDS bytes its own async ops wrote, so per-wave ASYNCcnt waits
  // are the only synchronization the jitter stream needs (no barriers).
  if (valid) {
    const unsigned l0 = (unsigned)(size_t)&jbuf[0][t * PT];
    const unsigned l1 = (unsigned)(size_t)&jbuf[1][t * PT];
    const float* g0 = jit + ((size_t)b * NH + 0) * T_LEN + e;
    const float* g1 = jit + ((size_t)b * NH + 1) * T_LEN + e;
    async_load_b128(l0,      g0);
    async_load_b128(l0 + 16, g0 + 4);
    async_load_b128(l1,      g1);
    async_load_b128(l1 + 16, g1 + 4);
  }
  __syncthreads();

  // Matrix-pipe scan of the 256 thread-totals by wave 0:
  // X[k][n] = totals[n*16+k];  D = L(16x16 lower-tri ones) * X  via 4 chained
  // V_WMMA_F32_16X16X4_F32 ops => D[m][n] = inclusive prefix within group n.
  if (t < 32) {
    const int hi = (t >= 16) ? 1 : 0;
    const int lm = t & 15;
    float Breg[8];
    #pragma unroll
    for (int r = 0; r < 8; ++r) Breg[r] = totals[lm * 16 + 2 * r + hi];  // B striping
    v8f acc = {};
    #pragma unroll
    for (int s = 0; s < 4; ++s) {
      v2f a, bb;
      // A (16x4) documented layout: M=lane&15, K = j + 2*(lane>=16); slice K+=4s
      a[0] = ((4 * s + 2 * hi    ) <= lm) ? 1.f : 0.f;
      a[1] = ((4 * s + 2 * hi + 1) <= lm) ? 1.f : 0.f;
      bb[0] = Breg[2 * s];
      bb[1] = Breg[2 * s + 1];
      acc = __builtin_amdgcn_wmma_f32_16x16x4_f32(false, a, false, bb,
                                                  (short)0, acc, false, false);
    }
    // D layout: lane holds column N=lane&15, rows M = r + 8*(lane>=16)
    #pragma unroll
    for (int r = 0; r < 8; ++r) scanD[lm * 16 + r + 8 * hi] = acc[r];
  }
  __syncthreads();
  if (t == 0) {       // exclusive scan of the 16 column totals (row 15 of D)
    float a = 0.f;
    #pragma unroll
    for (int n = 0; n < 16; ++n) { colOff[n] = a; a += scanD[n * 16 + 15]; }
  }
  __syncthreads();

  // Global inclusive cumsum of raw f0 for this thread's 8 elements.
  const float base = coff[b * NCHUNK + c] + (scanD[t] - totals[t]) + colOff[t >> 4];
  #pragma unroll
  for (int i = 0; i < PT; ++i) v[i] += base;

  // Harmonic loop: 2-deep double-buffered async jitter stream, no barriers.
  const size_t outB = (size_t)b * NH * T_LEN;
  #pragma unroll
  for (int h = 0; h < NH; ++h) {
    const int cur = h & 1;
    // Ops complete in order per wave: leaving <=2 in flight means harmonic h
    // (issued two issue-sites ago) has fully landed in jbuf[cur].
    if (h < NH - 1) asm volatile("s_wait_asynccnt 2" ::: "memory");
    else            asm volatile("s_wait_asynccnt 0" ::: "memory");

    float4 j0 = make_float4(0.f, 0.f, 0.f, 0.f), j1 = j0;
    if (valid) {
      const float4* jb = (const float4*)&jbuf[cur][t * PT];
      j0 = jb[0]; j1 = jb[1];
    }
    // Ensure the LDS reads above have completed before the async engine may
    // overwrite this buffer with harmonic h+2 (write-after-read on jbuf[cur]).
    asm volatile("s_wait_dscnt 0" ::: "memory");
    if (h + 2 < NH && valid) {
      const float* gn = jit + ((size_t)b * NH + (h + 2)) * T_LEN + e;
      const unsigned ln = (unsigned)(size_t)&jbuf[cur][t * PT];
      async_load_b128(ln,      gn);
      async_load_b128(ln + 16, gn + 4);
    }

    if (valid) {
      const float w = (TWO_PI_F / SR_F) * (float)(h + 1);
      float o[PT];
      o[0] = __sinf(fmaf(w, v[0], j0.x));
      o[1] = __sinf(fmaf(w, v[1], j0.y));
      o[2] = __sinf(fmaf(w, v[2], j0.z));
      o[3] = __sinf(fmaf(w, v[3], j0.w));
      o[4] = __sinf(fmaf(w, v[4], j1.x));
      o[5] = __sinf(fmaf(w, v[5], j1.y));
      o[6] = __sinf(fmaf(w, v[6], j1.z));
      o[7] = __sinf(fmaf(w, v[7], j1.w));
      float4* op = (float4*)(out + outB + (size_t)h * T_LEN + e);
      op[0] = make_float4(o[0], o[1], o[2], o[3]);
      op[1] = make_float4(o[4], o[5], o[6], o[7]);
    }
  }
}

extern "C" void kernel_launch(void* const* d_in, const int* in_sizes, int n_in,
                              void* d_out, int out_size, void* d_ws, size_t ws_size,
                              hipStream_t stream) {
  const float* f0  = (const float*)d_in[0];   // [8, 1, 240000]
  const float* jit = (const float*)d_in[1];   // [8, 9, 240000]
  float* out  = (float*)d_out;                // [8, 9, 240000]
  float* sums = (float*)d_ws;                 // NB*NCHUNK floats (3776 B)

  dim3 grid(NCHUNK, NB);
  k_chunk_sums<<<grid, THREADS, 0, stream>>>(f0, sums);
  k_scan<<<1, NB * 32, 0, stream>>>(sums);
  k_main<<<grid, THREADS, 0, stream>>>(f0, jit, sums, out);
}